// VariationalGCNEncoder_23587960389966
// MI455X (gfx1250) — compile-verified
//
#include <hip/hip_runtime.h>

#define N_NODES 100000
#define N_EDGES 3200000LL
#define DIM_IN  128
#define H_DIM   32
#define OUT_DIM 16
#define N_TILES (N_NODES / 16)   // 6250, exact

typedef __attribute__((ext_vector_type(2))) float v2f;
typedef __attribute__((ext_vector_type(8))) float v8f;

// ---------------- degree / normalization ----------------

__global__ void deg_kernel(const long long* __restrict__ ei, float* __restrict__ deg) {
  long long e = blockIdx.x * 256LL + threadIdx.x;
  if (e < N_EDGES) {
    int d = (int)ei[N_EDGES + e];
    atomicAdd(&deg[d], 1.0f);
  }
}

__global__ void dinv_kernel(float* __restrict__ deg) {
  int i = blockIdx.x * 256 + threadIdx.x;
  if (i < N_NODES) deg[i] = rsqrtf(deg[i] + 1.0f);   // +1 = self loop
}

// ---------------- GEMM1: Hpre = X @ W1  (fp32 WMMA 16x16x4) ----------------

__global__ void gemm_x_w1(const float* __restrict__ X, const float* __restrict__ W,
                          float* __restrict__ Hp) {
  const int wave = blockIdx.x * 4 + (threadIdx.x >> 5);
  if (wave >= N_TILES) return;                 // wave-uniform exit
  const int lane = threadIdx.x & 31;
  const int row0 = wave * 16;
  const int nn   = lane & 15;                  // M for A loads, N for B/C
  const int kk   = (lane >> 4) << 1;           // 0 or 2
  const float* xr = X + (long long)(row0 + nn) * DIM_IN;
  v8f acc0 = {}; v8f acc1 = {};
#pragma unroll
  for (int k0 = 0; k0 < DIM_IN; k0 += 4) {
    v2f a;  a[0]  = xr[k0 + kk];        a[1]  = xr[k0 + kk + 1];
    const float* wb = W + (k0 + kk) * H_DIM;
    v2f b0; b0[0] = wb[nn];             b0[1] = wb[H_DIM + nn];
    v2f bb; bb[0] = wb[16 + nn];        bb[1] = wb[H_DIM + 16 + nn];
    acc0 = __builtin_amdgcn_wmma_f32_16x16x4_f32(false, a, false, b0, (short)0, acc0, false, false);
    acc1 = __builtin_amdgcn_wmma_f32_16x16x4_f32(false, a, false, bb, (short)0, acc1, false, false);
  }
  const int mrow = row0 + ((lane >> 4) << 3);  // rows r or r+8
#pragma unroll
  for (int r = 0; r < 8; ++r) {
    Hp[(long long)(mrow + r) * H_DIM + nn]      = acc0[r];
    Hp[(long long)(mrow + r) * H_DIM + 16 + nn] = acc1[r];
  }
}

// ---------------- edge propagation: one wave per edge, lane = channel ----------------

__global__ void propagate(const long long* __restrict__ ei, const float* __restrict__ dinv,
                          const float* __restrict__ fin, float* __restrict__ fout) {
  long long t = blockIdx.x * 256LL + threadIdx.x;
  long long e = t >> 5;
  if (e >= N_EDGES) return;
  int c = (int)(t & 31);
  int s = (int)ei[e];
  int d = (int)ei[N_EDGES + e];
  float w = dinv[s] * dinv[d];
  atomicAdd(&fout[(long long)d * H_DIM + c], w * fin[(long long)s * H_DIM + c]);
}

// ---------------- layer-1 epilogue: self-loop + bias + relu (in place into agg) ----------------

__global__ void bias_relu_selfloop(float* __restrict__ agg, const float* __restrict__ hp,
                                   const float* __restrict__ dinv, const float* __restrict__ bias) {
  long long t = blockIdx.x * 256LL + threadIdx.x;
  if (t >= (long long)N_NODES * H_DIM) return;
  int i = (int)(t >> 5);
  int c = (int)(t & 31);
  float di = dinv[i];
  float v = agg[t] + di * di * hp[t] + bias[c];
  agg[t] = fmaxf(v, 0.0f);
}

// ---------------- layer-2 self-loop fold: agg2 += dinv^2 * h ----------------

__global__ void selfloop_add(float* __restrict__ agg, const float* __restrict__ h,
                             const float* __restrict__ dinv) {
  long long t = blockIdx.x * 256LL + threadIdx.x;
  if (t >= (long long)N_NODES * H_DIM) return;
  int i = (int)(t >> 5);
  float di = dinv[i];
  agg[t] += di * di * h[t];
}

// ---------------- pack Wc = [Wm | Wv | zeros] : 32x32 ----------------

__global__ void pack_wc(const float* __restrict__ Wm, const float* __restrict__ Wv,
                        float* __restrict__ Wc) {
  int t = blockIdx.x * 256 + threadIdx.x;
  if (t >= 32 * 32) return;
  int k = t >> 5, j = t & 31;
  float v = 0.0f;
  if (j < OUT_DIM)       v = Wm[k * OUT_DIM + j];
  else if (j == OUT_DIM) v = Wv[k];
  Wc[t] = v;
}

// ---------------- GEMM2: [mean_pre | var_pre] = Agg @ Wc  (fp32 WMMA) ----------------

__global__ void gemm_h_wc(const float* __restrict__ A, const float* __restrict__ Wc,
                          float* __restrict__ outMean, float* __restrict__ outVar) {
  const int wave = blockIdx.x * 4 + (threadIdx.x >> 5);
  if (wave >= N_TILES) return;
  const int lane = threadIdx.x & 31;
  const int row0 = wave * 16;
  const int nn   = lane & 15;
  const int kk   = (lane >> 4) << 1;
  const float* ar = A + (long long)(row0 + nn) * H_DIM;
  v8f acc0 = {}; v8f acc1 = {};
#pragma unroll
  for (int k0 = 0; k0 < H_DIM; k0 += 4) {
    v2f a;  a[0]  = ar[k0 + kk];   a[1]  = ar[k0 + kk + 1];
    const float* wb = Wc + (k0 + kk) * 32;
    v2f b0; b0[0] = wb[nn];        b0[1] = wb[32 + nn];
    v2f bb; bb[0] = wb[16 + nn];   bb[1] = wb[48 + nn];
    acc0 = __builtin_amdgcn_wmma_f32_16x16x4_f32(false, a, false, b0, (short)0, acc0, false, false);
    acc1 = __builtin_amdgcn_wmma_f32_16x16x4_f32(false, a, false, bb, (short)0, acc1, false, false);
  }
  const int mrow = row0 + ((lane >> 4) << 3);
#pragma unroll
  for (int r = 0; r < 8; ++r) {
    outMean[(long long)(mrow + r) * OUT_DIM + nn] = acc0[r];
    if (nn == 0) outVar[mrow + r] = acc1[r];   // col 16 of Wc == Wv
  }
}

// ---------------- finish: add biases, L2-normalize mean, softplus var (+1) ----------------

__global__ void finish_kernel(float* __restrict__ out, const float* __restrict__ bm,
                              const float* __restrict__ bv) {
  int i = blockIdx.x * 256 + threadIdx.x;
  if (i >= N_NODES) return;
  float* mp = out + (long long)i * OUT_DIM;
  float v[OUT_DIM];
  float ss = 0.0f;
#pragma unroll
  for (int j = 0; j < OUT_DIM; ++j) { v[j] = mp[j] + bm[j]; ss += v[j] * v[j]; }
  float inv = rsqrtf(ss);
#pragma unroll
  for (int j = 0; j < OUT_DIM; ++j) mp[j] = v[j] * inv;
  long long vi = (long long)N_NODES * OUT_DIM + i;
  float vp = out[vi] + bv[0];
  float sp = fmaxf(vp, 0.0f) + log1pf(expf(-fabsf(vp)));   // stable softplus
  out[vi] = sp + 1.0f;
}

// ---------------- launch ----------------

extern "C" void kernel_launch(void* const* d_in, const int* in_sizes, int n_in,
                              void* d_out, int out_size, void* d_ws, size_t ws_size,
                              hipStream_t stream) {
  (void)in_sizes; (void)n_in; (void)out_size; (void)ws_size;
  const float*     x  = (const float*)d_in[0];
  const long long* ei = (const long long*)d_in[1];   // int64 per reference
  const float*     W1 = (const float*)d_in[2];
  const float*     b1 = (const float*)d_in[3];
  const float*     Wm = (const float*)d_in[4];
  const float*     bm = (const float*)d_in[5];
  const float*     Wv = (const float*)d_in[6];
  const float*     bv = (const float*)d_in[7];
  float* out = (float*)d_out;

  float* dinv = (float*)d_ws;                              // N
  float* hpre = dinv + N_NODES;                            // N*32
  float* agg1 = hpre + (long long)N_NODES * H_DIM;         // N*32 (becomes h)
  float* agg2 = agg1 + (long long)N_NODES * H_DIM;         // N*32
  float* Wc   = agg2 + (long long)N_NODES * H_DIM;         // 32*32

  hipMemsetAsync(dinv, 0, (size_t)N_NODES * 4, stream);
  hipMemsetAsync(agg1, 0, (size_t)N_NODES * H_DIM * 4, stream);
  hipMemsetAsync(agg2, 0, (size_t)N_NODES * H_DIM * 4, stream);

  const int eb  = (int)((N_EDGES + 255) / 256);            // 12500
  const int nb  = (N_NODES + 255) / 256;                   // 391
  const int gb  = (N_TILES + 3) / 4;                       // 1563 blocks of 4 waves
  const long long lanesE = N_EDGES * 32;
  const int pb  = (int)((lanesE + 255) / 256);             // 400000
  const int fb  = (int)(((long long)N_NODES * H_DIM + 255) / 256);

  deg_kernel      <<<eb, 256, 0, stream>>>(ei, dinv);
  dinv_kernel     <<<nb, 256, 0, stream>>>(dinv);
  gemm_x_w1       <<<gb, 128, 0, stream>>>(x, W1, hpre);
  propagate       <<<pb, 256, 0, stream>>>(ei, dinv, hpre, agg1);
  bias_relu_selfloop<<<fb, 256, 0, stream>>>(agg1, hpre, dinv, b1);
  propagate       <<<pb, 256, 0, stream>>>(ei, dinv, agg1, agg2);
  selfloop_add    <<<fb, 256, 0, stream>>>(agg2, agg1, dinv);
  pack_wc         <<<4, 256, 0, stream>>>(Wm, Wv, Wc);
  gemm_h_wc       <<<gb, 128, 0, stream>>>(agg2, Wc, out, out + (long long)N_NODES * OUT_DIM);
  finish_kernel   <<<nb, 256, 0, stream>>>(out, bm, bv);
}